// HybridSelfAttention_65481071408940
// MI455X (gfx1250) — compile-verified
//
#include <hip/hip_runtime.h>
#include <math.h>

typedef __attribute__((ext_vector_type(2))) float v2f;
typedef __attribute__((ext_vector_type(8))) float v8f;

constexpr int E = 512;    // embed
constexpr int S = 1024;   // seq
constexpr int BB = 4;     // batch
constexpr int H = 16;     // hidden

// Hardware tanh (CDNA5 V_TANH_F32 transcendental). Builtin preferred so the
// compiler handles the TRANS-op hazard; asm fallback carries its own v_nop.
#if defined(__has_builtin) && __has_builtin(__builtin_amdgcn_tanhf)
__device__ __forceinline__ float fast_tanh(float x) { return __builtin_amdgcn_tanhf(x); }
#else
__device__ __forceinline__ float fast_tanh(float x) {
  float r;
  asm volatile("v_tanh_f32 %0, %1\n\tv_nop" : "=v"(r) : "v"(x));
  return r;
}
#endif

// ---------------------------------------------------------------------------
// Kernel 1: combined weights  CWq[h,e] = sum_f W1[h,f]   * Wq[f,e]
//                             CWk[h,e] = sum_f W1[h,E+f] * Wk[f,e]
// M=16 (h), N=512 (e), K=512 (f).  One wave per 16-wide e-tile.
// ---------------------------------------------------------------------------
__global__ __launch_bounds__(32) void cw_kernel(const float* __restrict__ W1,
                                                const float* __restrict__ Wq,
                                                const float* __restrict__ Wk,
                                                float* __restrict__ CWq,
                                                float* __restrict__ CWk) {
  const int lane = threadIdx.x;
  const int n0   = blockIdx.x * 16;
  const int half = lane >> 4;       // lane half selects K pair {0,1} vs {2,3}
  const int l    = lane & 15;
  v8f accq = {}; v8f acck = {};
  for (int k = 0; k < E; k += 4) {
    const int ka = k + 2 * half;
    // A-frag: row m=l of W1 (row stride 2E), contiguous K pair
    v2f aq = *(const v2f*)(W1 + l * (2 * E) + ka);
    v2f ak = *(const v2f*)(W1 + l * (2 * E) + E + ka);
    // B-frag: B[k=f][n=e] = Wq[f*E + e]  (stride-E gather across the K pair)
    v2f bq; bq.x = Wq[ka * E + n0 + l]; bq.y = Wq[(ka + 1) * E + n0 + l];
    v2f bk; bk.x = Wk[ka * E + n0 + l]; bk.y = Wk[(ka + 1) * E + n0 + l];
    accq = __builtin_amdgcn_wmma_f32_16x16x4_f32(false, aq, false, bq, (short)0, accq, false, false);
    acck = __builtin_amdgcn_wmma_f32_16x16x4_f32(false, ak, false, bk, (short)0, acck, false, false);
  }
#pragma unroll
  for (int v = 0; v < 8; ++v) {
    const int m = v + 8 * half;              // h
    CWq[m * E + n0 + l] = accq[v];
    CWk[m * E + n0 + l] = acck[v];
  }
}

// ---------------------------------------------------------------------------
// Kernel 2: A[m,h] = sum_e x[m,e]*CWq[h,e] + b1[h];  Bv[m,h] = sum_e x[m,e]*CWk[h,e]
// M=4096 rows, N=16 (h), K=512 (e). One wave per 16-row tile; A-frag shared.
// ---------------------------------------------------------------------------
__global__ __launch_bounds__(32) void ab_kernel(const float* __restrict__ x,
                                                const float* __restrict__ CWq,
                                                const float* __restrict__ CWk,
                                                const float* __restrict__ b1,
                                                float* __restrict__ Avec,
                                                float* __restrict__ Bvec) {
  const int lane = threadIdx.x;
  const int m0   = blockIdx.x * 16;
  const int half = lane >> 4;
  const int l    = lane & 15;
  v8f accA = {}; v8f accB = {};
  const float* xrow = x + (size_t)(m0 + l) * E;
  const float* cq   = CWq + l * E;   // column n=h=l  ->  CW row l, contiguous in e
  const float* ck   = CWk + l * E;
  for (int k = 0; k < E; k += 4) {
    const int ka = k + 2 * half;
    v2f a  = *(const v2f*)(xrow + ka);
    v2f bq = *(const v2f*)(cq + ka);
    v2f bk = *(const v2f*)(ck + ka);
    accA = __builtin_amdgcn_wmma_f32_16x16x4_f32(false, a, false, bq, (short)0, accA, false, false);
    accB = __builtin_amdgcn_wmma_f32_16x16x4_f32(false, a, false, bk, (short)0, accB, false, false);
  }
  const float bias = b1[l];
#pragma unroll
  for (int v = 0; v < 8; ++v) {
    const int m = m0 + v + 8 * half;
    Avec[m * H + l] = accA[v] + bias;
    Bvec[m * H + l] = accB[v];
  }
}

// ---------------------------------------------------------------------------
// Kernel 3: V projection  V[m,f] = sum_e x[m,e] * Wv[f,e]
// M=4096, N=512, K=512.  Both A and B frags are contiguous float2 loads.
// ---------------------------------------------------------------------------
__global__ __launch_bounds__(32) void vproj_kernel(const float* __restrict__ x,
                                                   const float* __restrict__ Wv,
                                                   float* __restrict__ V) {
  const int lane = threadIdx.x;
  const int m0   = blockIdx.x * 16;
  const int n0   = blockIdx.y * 16;
  const int half = lane >> 4;
  const int l    = lane & 15;
  v8f acc = {};
  const float* xrow = x  + (size_t)(m0 + l) * E;
  const float* wrow = Wv + (size_t)(n0 + l) * E;   // B[k=e][n=f] = Wv[f][e]
  for (int k = 0; k < E; k += 4) {
    const int ka = k + 2 * half;
    v2f a = *(const v2f*)(xrow + ka);
    v2f b = *(const v2f*)(wrow + ka);
    acc = __builtin_amdgcn_wmma_f32_16x16x4_f32(false, a, false, b, (short)0, acc, false, false);
  }
#pragma unroll
  for (int v = 0; v < 8; ++v)
    V[(size_t)(m0 + v + 8 * half) * E + n0 + l] = acc[v];
}

// ---------------------------------------------------------------------------
// Kernel 4: fused tanh-net attention.
// One 256-thread workgroup (8 waves) per (batch, 16-row i-tile).
// Per 16-wide j-step: stage the B tile in LDS, 256 threads each compute one
// exp(logit) via hardware V_TANH_F32 (logits bounded by sum|w2| -> no max
// subtraction needed), shuffle-reduce row denominators, then each wave WMMAs
// the shared P tile against its private 64 V-columns. Next j-step's V tile
// is prefetched (global_prefetch_b8).
// ---------------------------------------------------------------------------
__global__ __launch_bounds__(256) void attn_kernel(const float* __restrict__ Avec,
                                                   const float* __restrict__ Bvec,
                                                   const float* __restrict__ V,
                                                   const float* __restrict__ w2,
                                                   const float* __restrict__ b2,
                                                   float* __restrict__ out) {
  __shared__ float sA[16 * H];
  __shared__ float sB[16 * H];
  __shared__ float sW2[H];
  __shared__ float sP[16 * 16];
  __shared__ float sDen[16];

  const int tid  = threadIdx.x;
  const int lane = tid & 31;
  const int wave = tid >> 5;        // 0..7, owns 64 output columns
  const int half = lane >> 4;
  const int l    = lane & 15;

  const int tileIdx = blockIdx.x;              // over BB*S/16
  const int b       = tileIdx / (S / 16);
  const int i0      = (tileIdx % (S / 16)) * 16;
  const int rowbase = b * S + i0;

  if (tid < 16 * H) sA[tid]  = Avec[(rowbase + (tid >> 4)) * H + (tid & 15)];
  if (tid < H)      sW2[tid] = w2[tid];
  if (tid < 16)     sDen[tid] = 0.0f;

  const float bias2 = b2[0];
  const int   il = tid >> 4;        // logit row this thread computes
  const int   jl = tid & 15;        // logit col this thread computes

  v8f acc[4] = {};

  for (int j0 = 0; j0 < S; j0 += 16) {
    // ---- 0) stage B tile (16 rows x 16 h) in LDS; 1 load per thread -------
    sB[tid] = Bvec[(size_t)(b * S + j0 + (tid >> 4)) * H + (tid & 15)];
    // prefetch next j-step's V slice for this wave (16 rows x 2 addresses per wave)
    if (j0 + 16 < S)
      __builtin_prefetch(V + (size_t)(b * S + j0 + 16 + l) * E + wave * 64 + half * 32, 0, 1);
    __syncthreads();   // sB ready; also orders prev-step WMMA reads vs sP write

    // ---- 1) exp(logit) tile: one element per thread -----------------------
    float logit = bias2;
#pragma unroll
    for (int h = 0; h < H; ++h)
      logit += sW2[h] * fast_tanh(sA[il * H + h] + sB[jl * H + h]);
    float ev = __expf(logit);
    sP[il * 16 + jl] = ev;
    // row-denominator: reduce the 16 j-lanes (they share a 16-lane half)
#pragma unroll
    for (int off = 8; off; off >>= 1)
      ev += __shfl_xor(ev, off, 16);
    if (jl == 0) sDen[il] += ev;    // unique writer per row
    __syncthreads();

    // ---- 2) acc(16x64) += P(16x16) x V(16x64) via f32 WMMA ----------------
#pragma unroll
    for (int k = 0; k < 16; k += 4) {
      const int ka = k + 2 * half;
      v2f a = *(const v2f*)(&sP[l * 16 + ka]);
      const float* vb = V + (size_t)(b * S + j0 + ka) * E;   // rows ka, ka+1
#pragma unroll
      for (int t = 0; t < 4; ++t) {
        const int n = wave * 64 + t * 16 + l;
        v2f bf; bf.x = vb[n]; bf.y = vb[E + n];
        acc[t] = __builtin_amdgcn_wmma_f32_16x16x4_f32(false, a, false, bf, (short)0, acc[t], false, false);
      }
    }
  }

  // ---- normalize by softmax denominator and store --------------------------
  // (sDen writes of the last step precede that step's mid-loop barrier)
#pragma unroll
  for (int v = 0; v < 8; ++v) {
    const int m   = v + 8 * half;
    const float r = 1.0f / sDen[m];
#pragma unroll
    for (int t = 0; t < 4; ++t) {
      const int n = wave * 64 + t * 16 + l;
      out[(size_t)(rowbase + m) * E + n] = acc[t][v] * r;
    }
  }
}

// ---------------------------------------------------------------------------
extern "C" void kernel_launch(void* const* d_in, const int* in_sizes, int n_in,
                              void* d_out, int out_size, void* d_ws, size_t ws_size,
                              hipStream_t stream) {
  const float* x  = (const float*)d_in[0];
  const float* Wq = (const float*)d_in[1];
  const float* Wk = (const float*)d_in[2];
  const float* Wv = (const float*)d_in[3];
  const float* W1 = (const float*)d_in[4];
  const float* b1 = (const float*)d_in[5];
  const float* w2 = (const float*)d_in[6];
  const float* b2 = (const float*)d_in[7];
  float* out = (float*)d_out;

  float* ws  = (float*)d_ws;
  float* CWq = ws;                          // 16*512
  float* CWk = ws + 8192;                   // 16*512
  float* Av  = ws + 16384;                  // 4096*16
  float* Bv  = ws + 16384 + 65536;          // 4096*16
  float* V   = ws + 16384 + 2 * 65536;      // 4096*512

  cw_kernel  <<<dim3(E / 16),               dim3(32),  0, stream>>>(W1, Wq, Wk, CWq, CWk);
  ab_kernel  <<<dim3(BB * S / 16),          dim3(32),  0, stream>>>(x, CWq, CWk, b1, Av, Bv);
  vproj_kernel<<<dim3(BB * S / 16, E / 16), dim3(32),  0, stream>>>(x, Wv, V);
  attn_kernel<<<dim3(BB * S / 16),          dim3(256), 0, stream>>>(Av, Bv, V, w2, b2, out);
}